// HeteroGNN_62886911148643
// MI455X (gfx1250) — compile-verified
//
#include <hip/hip_runtime.h>
#include <math.h>

typedef __attribute__((ext_vector_type(2))) float v2f;
typedef __attribute__((ext_vector_type(8))) float v8f;

#define NNODES 50000
#define NEDGE  500000
#define HDIM   128
#define EDIM   16
#define INCH   144   // HDIM + EDIM
#define ATTND  64

// ---------------------------------------------------------------------------
// Scatter node features: agg[dst, 0:128] += h[src, 0:128]  (one thread = 4 ch)
// ---------------------------------------------------------------------------
__global__ void scatter_node_feats(const float* __restrict__ h,
                                   const int* __restrict__ ei,
                                   float* __restrict__ agg, int E) {
  int t = blockIdx.x * blockDim.x + threadIdx.x;   // E*32 threads exactly
  int e = t >> 5;
  int q = (t & 31) * 4;
  int src = ei[e];
  int dst = ei[E + e];
  const float4 v = *(const float4*)(h + (long)src * HDIM + q);
  float* base = agg + (long)dst * INCH + q;
  atomicAdd(base + 0, v.x);
  atomicAdd(base + 1, v.y);
  atomicAdd(base + 2, v.z);
  atomicAdd(base + 3, v.w);
}

// ---------------------------------------------------------------------------
// Scatter edge features: agg[dst, 128:144] += ef[e, 0:16]
// ---------------------------------------------------------------------------
__global__ void scatter_edge_feats(const float* __restrict__ ef,
                                   const int* __restrict__ ei,
                                   float* __restrict__ agg, int E) {
  int t = blockIdx.x * blockDim.x + threadIdx.x;
  int e = t >> 2;
  if (e >= E) return;
  int q = (t & 3) * 4;
  int dst = ei[E + e];
  const float4 v = *(const float4*)(ef + (long)e * EDIM + q);
  float* base = agg + (long)dst * INCH + HDIM + q;
  atomicAdd(base + 0, v.x);
  atomicAdd(base + 1, v.y);
  atomicAdd(base + 2, v.z);
  atomicAdd(base + 3, v.w);
}

// ---------------------------------------------------------------------------
// C[M x 128] = A[M x K](lda) @ B[K x 128] + bias,  f32 WMMA 16x16x4.
// One wave per 16x16 output tile; grid is exact -> EXEC all-ones for WMMA.
// A-frag (16x4 f32): lanes 0-15 carry M rows, half-wave selects K pair.
// C/D: 8 VGPRs, lane<16 -> M=r, lane>=16 -> M=r+8.
// ---------------------------------------------------------------------------
__global__ void gemm_bias_wmma(const float* __restrict__ A, int lda, int K,
                               const float* __restrict__ B,
                               const float* __restrict__ bias,
                               float* __restrict__ C) {
  const int wavesPerBlock = blockDim.x >> 5;
  int wave = blockIdx.x * wavesPerBlock + (threadIdx.x >> 5);
  int lane = threadIdx.x & 31;
  const int colTiles = HDIM / 16;            // 8
  int rowBase = (wave / colTiles) * 16;
  int colBase = (wave % colTiles) * 16;
  int half = lane >> 4;                      // K-pair select
  int m = lane & 15;

  v8f c = {};
  const float* arow = A + (long)(rowBase + m) * lda;
  const float* bcol = B + colBase + m;
  for (int k = 0; k < K; k += 4) {
    int ka = k + 2 * half;
    v2f a, b;
    a.x = arow[ka];
    a.y = arow[ka + 1];
    b.x = bcol[(long)ka * HDIM];
    b.y = bcol[(long)(ka + 1) * HDIM];
    c = __builtin_amdgcn_wmma_f32_16x16x4_f32(false, a, false, b,
                                              (short)0, c, false, false);
  }
  float bv = bias[colBase + m];
#pragma unroll
  for (int r = 0; r < 8; ++r) {
    int row = rowBase + r + 8 * half;
    C[(long)row * HDIM + colBase + m] = c[r] + bv;
  }
}

// ---------------------------------------------------------------------------
// z contribution: sum_n dot(tanh(emb[n] @ Wa1 + ba1), Wa2) -> atomic into zsum
// block = 4 nodes x 64 attn channels
// ---------------------------------------------------------------------------
__global__ void attn_z(const float* __restrict__ emb,
                       const float* __restrict__ Wa1,   // [HDIM x ATTND]
                       const float* __restrict__ ba1,
                       const float* __restrict__ Wa2,
                       float* __restrict__ zsum) {
  __shared__ float semb[4][HDIM];
  __shared__ float sred[256];
  int nn = threadIdx.x >> 6;
  int j = threadIdx.x & 63;
  long node0 = (long)blockIdx.x * 4;
  for (int i = threadIdx.x; i < 4 * HDIM; i += blockDim.x)
    semb[i >> 7][i & 127] = emb[node0 * HDIM + i];
  __syncthreads();
  float s = 0.f;
#pragma unroll 4
  for (int k = 0; k < HDIM; ++k)
    s += semb[nn][k] * Wa1[k * ATTND + j];
  sred[threadIdx.x] = tanhf(s + ba1[j]) * Wa2[j];
  __syncthreads();
  for (int off = 32; off > 0; off >>= 1) {
    if (j < off) sred[threadIdx.x] += sred[threadIdx.x + off];
    __syncthreads();
  }
  if (j == 0) atomicAdd(zsum, sred[threadIdx.x]);
}

// ---------------------------------------------------------------------------
// mix = softmax(z/N) weighted sum (in-place over emb_a) + BN column sums
// ---------------------------------------------------------------------------
__global__ void combine_stats(float* __restrict__ emb_a,
                              const float* __restrict__ emb_b,
                              const float* __restrict__ zsum,
                              float* __restrict__ colsum,
                              float* __restrict__ colsumsq) {
  float za = zsum[0] * (1.0f / NNODES);
  float zb = zsum[1] * (1.0f / NNODES);
  float mx = fmaxf(za, zb);
  float ea = expf(za - mx), eb = expf(zb - mx);
  float inv = 1.0f / (ea + eb);
  float aA = ea * inv, aB = eb * inv;

  int tid = blockIdx.x * blockDim.x + threadIdx.x;
  int nthreads = gridDim.x * blockDim.x;
  int c = tid & 127;
  int stride = nthreads >> 7;
  float s = 0.f, s2 = 0.f;
  for (int row = tid >> 7; row < NNODES; row += stride) {
    long idx = (long)row * HDIM + c;
    float v = aA * emb_a[idx] + aB * emb_b[idx];
    emb_a[idx] = v;
    s += v;
    s2 += v * v;
  }
  atomicAdd(&colsum[c], s);
  atomicAdd(&colsumsq[c], s2);
}

__global__ void bn_finalize(const float* __restrict__ colsum,
                            const float* __restrict__ colsumsq,
                            const float* __restrict__ gamma,
                            const float* __restrict__ beta,
                            float* __restrict__ nscale,
                            float* __restrict__ nbias) {
  int c = threadIdx.x;
  float mean = colsum[c] * (1.0f / NNODES);
  float var = colsumsq[c] * (1.0f / NNODES) - mean * mean;
  float g = rsqrtf(var + 1.0f) * gamma[c];   // BN_EPS = 1
  nscale[c] = g;
  nbias[c] = beta[c] - mean * g;
}

__global__ void bn_apply_prelu(const float* __restrict__ hmix,
                               float* __restrict__ hout,
                               const float* __restrict__ nscale,
                               const float* __restrict__ nbias,
                               const float* __restrict__ prelu) {
  float a = prelu[0];
  long idx = (long)blockIdx.x * blockDim.x + threadIdx.x;
  int c = (int)(idx & 127);
  float v = hmix[idx] * nscale[c] + nbias[c];
  hout[idx] = v >= 0.f ? v : a * v;
}

// ---------------------------------------------------------------------------
// Edge head: sigmoid(dot(h[src], h[dst])) — one wave32 per edge.
// ---------------------------------------------------------------------------
__global__ void edge_head(const float* __restrict__ h,
                          const int* __restrict__ ei,
                          float* __restrict__ out, int E) {
  int t = blockIdx.x * blockDim.x + threadIdx.x;  // E*32 threads exactly
  int e = t >> 5;
  int lane = t & 31;
  int src = ei[e], dst = ei[E + e];
  const float* ps = h + (long)src * HDIM;
  const float* pd = h + (long)dst * HDIM;
  float s = 0.f;
#pragma unroll
  for (int k = lane; k < HDIM; k += 32) s += ps[k] * pd[k];
#pragma unroll
  for (int off = 16; off > 0; off >>= 1) s += __shfl_xor(s, off, 32);
  if (lane == 0) out[e] = 1.0f / (1.0f + expf(-s));
}

// ---------------------------------------------------------------------------
extern "C" void kernel_launch(void* const* d_in, const int* in_sizes, int n_in,
                              void* d_out, int out_size, void* d_ws, size_t ws_size,
                              hipStream_t stream) {
  const float* x    = (const float*)d_in[0];
  const int*   ei_a = (const int*)d_in[1];
  const int*   ei_b = (const int*)d_in[2];
  const float* ef_a = (const float*)d_in[3];
  const float* ef_b = (const float*)d_in[4];
  // params dict order: per layer l (10 each): W{l}a, b{l}a, W{l}b, b{l}b,
  //   Wa1_l, ba1_l, Wa2_l, gamma_l, beta_l, prelu_l ; then Wpost, bpost
  void* const* p = (void* const*)(d_in + 5);

  float* ws = (float*)d_ws;
  float* agg_a = ws;    ws += (size_t)NNODES * INCH;
  float* agg_b = ws;    ws += (size_t)NNODES * INCH;
  float* emb_a = ws;    ws += (size_t)NNODES * HDIM;   // also mix (in-place)
  float* emb_b = ws;    ws += (size_t)NNODES * HDIM;   // also hpost (reused)
  float* h1    = ws;    ws += (size_t)NNODES * HDIM;
  float* zsum  = ws;    ws += 2;
  float* colsum   = ws; ws += HDIM;
  float* colsumsq = ws; ws += HDIM;
  float* nscale   = ws; ws += HDIM;
  float* nbias    = ws; ws += HDIM;

  const int scatterNodeBlocks = NEDGE * 32 / 256;            // 62500
  const int scatterEdgeBlocks = (NEDGE * 4 + 255) / 256;     // 7813
  const int gemmBlocks = (NNODES / 16) * (HDIM / 16) / 8;    // 3125 (8 waves/blk)
  const int applyBlocks = NNODES * HDIM / 256;               // 25000
  const int edgeBlocks = NEDGE * 32 / 256;                   // 62500

  const float* hcur = x;
  for (int l = 0; l < 2; ++l) {
    const float* Wla   = (const float*)p[l * 10 + 0];
    const float* bla   = (const float*)p[l * 10 + 1];
    const float* Wlb   = (const float*)p[l * 10 + 2];
    const float* blb   = (const float*)p[l * 10 + 3];
    const float* Wa1   = (const float*)p[l * 10 + 4];
    const float* ba1   = (const float*)p[l * 10 + 5];
    const float* Wa2   = (const float*)p[l * 10 + 6];
    const float* gamma = (const float*)p[l * 10 + 7];
    const float* beta  = (const float*)p[l * 10 + 8];
    const float* prelu = (const float*)p[l * 10 + 9];

    hipMemsetAsync(agg_a, 0, (size_t)NNODES * INCH * sizeof(float), stream);
    hipMemsetAsync(agg_b, 0, (size_t)NNODES * INCH * sizeof(float), stream);
    hipMemsetAsync(zsum, 0, 2 * sizeof(float), stream);
    hipMemsetAsync(colsum, 0, HDIM * sizeof(float), stream);
    hipMemsetAsync(colsumsq, 0, HDIM * sizeof(float), stream);

    scatter_node_feats<<<scatterNodeBlocks, 256, 0, stream>>>(hcur, ei_a, agg_a, NEDGE);
    scatter_node_feats<<<scatterNodeBlocks, 256, 0, stream>>>(hcur, ei_b, agg_b, NEDGE);
    scatter_edge_feats<<<scatterEdgeBlocks, 256, 0, stream>>>(ef_a, ei_a, agg_a, NEDGE);
    scatter_edge_feats<<<scatterEdgeBlocks, 256, 0, stream>>>(ef_b, ei_b, agg_b, NEDGE);

    gemm_bias_wmma<<<gemmBlocks, 256, 0, stream>>>(agg_a, INCH, INCH, Wla, bla, emb_a);
    gemm_bias_wmma<<<gemmBlocks, 256, 0, stream>>>(agg_b, INCH, INCH, Wlb, blb, emb_b);

    attn_z<<<NNODES / 4, 256, 0, stream>>>(emb_a, Wa1, ba1, Wa2, zsum + 0);
    attn_z<<<NNODES / 4, 256, 0, stream>>>(emb_b, Wa1, ba1, Wa2, zsum + 1);

    combine_stats<<<1024, 256, 0, stream>>>(emb_a, emb_b, zsum, colsum, colsumsq);
    bn_finalize<<<1, HDIM, 0, stream>>>(colsum, colsumsq, gamma, beta, nscale, nbias);
    // layer-1 input (h1) already consumed by the scatter pass -> safe to overwrite
    bn_apply_prelu<<<applyBlocks, 256, 0, stream>>>(emb_a, h1, nscale, nbias, prelu);
    hcur = h1;
  }

  const float* Wpost = (const float*)p[20];
  const float* bpost = (const float*)p[21];
  float* hpost = emb_b;  // free after last combine
  gemm_bias_wmma<<<gemmBlocks, 256, 0, stream>>>(h1, HDIM, HDIM, Wpost, bpost, hpost);

  float* out = (float*)d_out;
  edge_head<<<edgeBlocks, 256, 0, stream>>>(hpost, ei_a, out, NEDGE);
  edge_head<<<edgeBlocks, 256, 0, stream>>>(hpost, ei_b, out + NEDGE, NEDGE);
}